// BatchMemoryAttentionLayer_44341242364180
// MI455X (gfx1250) — compile-verified
//
#include <hip/hip_runtime.h>
#include <hip/hip_bf16.h>
#include <float.h>

// ---- problem constants (match reference) ----
constexpr int Bb   = 32;
constexpr int Tt   = 128;
constexpr int Hh   = 768;
constexpr int Nn   = 1024;
constexpr int Mm   = 131072;
constexpr int DKk  = 128;
constexpr int DVv  = 512;
constexpr int NROWS = 512;
constexpr int VPR   = 256;   // Mm / NROWS
constexpr int KTOP  = 32;

typedef float v2f __attribute__((ext_vector_type(2)));
typedef float v8f __attribute__((ext_vector_type(8)));
typedef int   v4i __attribute__((ext_vector_type(4)));

// V_WMMA_F32_16X16X4_F32 : D[16x16] = A[16x4] * B[4x16] + C, all f32, wave32.
static __device__ inline v8f wmma4(v2f a, v2f b, v8f c) {
  return __builtin_amdgcn_wmma_f32_16x16x4_f32(
      /*neg_a=*/false, a, /*neg_b=*/false, b,
      /*c_mod=*/(short)0, c, /*reuse_a=*/false, /*reuse_b=*/false);
}

// gfx1250 async global->LDS copy (ASYNCcnt tracked), if the toolchain has it.
// Probe round 2: param0 is v4i addrspace(1)* (clang prints AS1 as __device__).
#if __has_builtin(__builtin_amdgcn_global_load_async_to_lds_b128) && \
    __has_builtin(__builtin_amdgcn_s_wait_asynccnt)
#define HAVE_ASYNC_LDS 1
typedef __attribute__((address_space(1))) v4i* g_v4i_p;
typedef __attribute__((address_space(3))) v4i* l_v4i_p;
static __device__ inline void async_copy16(const float* gsrc, float* ldst) {
  __builtin_amdgcn_global_load_async_to_lds_b128(
      (g_v4i_p)(v4i*)const_cast<float*>(gsrc),
      (l_v4i_p)(v4i*)ldst, 0, 0);
}
#else
#define HAVE_ASYNC_LDS 0
#endif

// ============================================================
// Kernel A: queries[N,128] = concat(start_enc,end_enc)[N,1536] @ qw + qb
// 64 blocks x 256 thr (8 waves). Wave w owns output cols [16w,16w+16).
// ============================================================
__global__ __launch_bounds__(256) void k_queries(
    const float* __restrict__ enc, const int* __restrict__ mbp,
    const int* __restrict__ msp, const int* __restrict__ mep,
    const float* __restrict__ qw, const float* __restrict__ qb,
    float* __restrict__ qout) {
  constexpr int KC = 256;                 // K-chunk of the 1536 reduction
  __shared__ float As[16][KC + 4];        // pad 4: (KC+4)%64==4 -> conflict free
  const int tid  = threadIdx.x;
  const int lane = tid & 31, wave = tid >> 5;
  const int qbase = blockIdx.x * 16;
  const int colbase = wave * 16;
  const int m    = lane & 15;             // A row / B col / D col (lane%16)
  const int koff = (lane >> 4) * 2;       // K half-wave split (ISA A layout)
  v8f acc = {};
  for (int kc = 0; kc < 2 * Hh; kc += KC) {
    for (int idx = tid; idx < 16 * KC; idx += 256) {
      int m2 = idx / KC;
      int f  = kc + (idx % KC);
      int men = qbase + m2;
      int bpos = mbp[men];
      int pos  = (f < Hh) ? msp[men] : mep[men];
      int hh   = (f < Hh) ? f : (f - Hh);
      As[m2][idx % KC] = enc[((long)bpos * Tt + pos) * Hh + hh];
    }
    __syncthreads();
    for (int k = 0; k < KC; k += 4) {
      v2f a, b;
      a.x = As[m][k + koff];
      a.y = As[m][k + koff + 1];
      const int kg = kc + k + koff;
      b.x = qw[(long)kg * DKk + colbase + m];
      b.y = qw[(long)(kg + 1) * DKk + colbase + m];
      acc = wmma4(a, b, acc);
    }
    __syncthreads();
  }
  const float bias = qb[colbase + m];
  for (int v = 0; v < 8; ++v) {
    int row = v + 8 * (lane >> 4);        // ISA C/D layout
    qout[(long)(qbase + row) * DKk + colbase + m] = acc[v] + bias;
  }
}

// ============================================================
// Kernel B: scores[q, slot] = queries . keys^T ; per (mention, table-row)
// reduce 256 slots -> (max, argmax).  Grid (N/16, 512), 256 thr (8 waves).
// Keys stay L2-resident (64MB < 192MB L2) -> compute bound on f32 WMMA.
// A-fragments are preloaded into 64 VGPRs (invariant across all chunks);
// key chunks stream via GLOBAL_LOAD_ASYNC_TO_LDS_B128 when available.
// ============================================================
__global__ __launch_bounds__(256) void k_scores(
    const float* __restrict__ qbuf, const float* __restrict__ keys,
    float* __restrict__ row_max, int* __restrict__ row_arg) {
  __shared__ float Qs[16][DKk + 4];   //  8448 B
  __shared__ float Ks[VPR][36];       // 36864 B (32-wide K chunk, pad->36; row=144B)
  __shared__ float Ss[16][264];       // 16896 B score tile
  __shared__ float Pv[16][16];        //  1024 B
  __shared__ int   Pi[16][16];        //  1024 B   (total 64256 < 64KB)
  const int tid  = threadIdx.x;
  const int lane = tid & 31, wave = tid >> 5;
  const int qbase = blockIdx.x * 16;
  const int row   = blockIdx.y;
  const long slot0 = (long)row * VPR;

  for (int idx = tid; idx < 16 * DKk; idx += 256)
    Qs[idx / DKk][idx % DKk] = qbuf[(long)(qbase + idx / DKk) * DKk + (idx % DKk)];
  __syncthreads();

  const int m    = lane & 15;
  const int koff = (lane >> 4) * 2;
  const int sb0  = wave * 32;         // 2 column tiles of 16 slots per wave
  const int sb1  = sb0 + 16;

  // Preload all 32 A-fragments (K=128) into registers once.
  v2f afrag[32];
#pragma unroll
  for (int f = 0; f < 32; ++f) {
    afrag[f].x = Qs[m][4 * f + koff];
    afrag[f].y = Qs[m][4 * f + koff + 1];
  }

  v8f acc0 = {}, acc1 = {};
  for (int kc = 0; kc < DKk; kc += 32) {
#if HAVE_ASYNC_LDS
    for (int u = tid; u < VPR * 8; u += 256) {       // 2048 x 16B async copies
      int s = u >> 3, j = (u & 7) * 4;
      async_copy16(&keys[(slot0 + s) * DKk + kc + j], &Ks[s][j]);
    }
    __builtin_amdgcn_s_wait_asynccnt(0);
#else
    for (int u = tid; u < VPR * 8; u += 256) {       // 2048 float4 loads
      int s = u >> 3, j = (u & 7) * 4;
      const float4 kv = *(const float4*)&keys[(slot0 + s) * DKk + kc + j];
      Ks[s][j + 0] = kv.x; Ks[s][j + 1] = kv.y;
      Ks[s][j + 2] = kv.z; Ks[s][j + 3] = kv.w;
    }
#endif
    __syncthreads();
    if (kc + 32 < DKk)                               // global_prefetch_b8
      __builtin_prefetch(&keys[(slot0 + tid) * DKk + kc + 32], 0, 1);
#pragma unroll
    for (int k = 0; k < 32; k += 4) {
      const v2f a = afrag[(kc >> 2) + (k >> 2)];
      v2f b0, b1;
      b0.x = Ks[sb0 + m][k + koff];     b0.y = Ks[sb0 + m][k + koff + 1];
      b1.x = Ks[sb1 + m][k + koff];     b1.y = Ks[sb1 + m][k + koff + 1];
      acc0 = wmma4(a, b0, acc0);
      acc1 = wmma4(a, b1, acc1);
    }
    __syncthreads();
  }
  for (int v = 0; v < 8; ++v) {
    int qq = v + 8 * (lane >> 4);
    Ss[qq][sb0 + m] = acc0[v];
    Ss[qq][sb1 + m] = acc1[v];
  }
  __syncthreads();
  {                                    // stage 1: 16 threads per mention
    int q = tid >> 4, part = tid & 15;
    float bv = -FLT_MAX; int bi = 0;
    for (int j = 0; j < 16; ++j) {
      int s = part * 16 + j;
      float x = Ss[q][s];
      if (x > bv) { bv = x; bi = s; }  // ascending scan => first-occurrence
    }
    Pv[q][part] = bv; Pi[q][part] = bi;
  }
  __syncthreads();
  if (tid < 16) {                      // stage 2
    float bv = -FLT_MAX; int bi = 0x7FFFFFFF;
    for (int p = 0; p < 16; ++p) {
      float x = Pv[tid][p]; int i = Pi[tid][p];
      if (x > bv || (x == bv && i < bi)) { bv = x; bi = i; }
    }
    row_max[(long)(qbase + tid) * NROWS + row] = bv;
    row_arg[(long)(qbase + tid) * NROWS + row] = bi;
  }
}

// ============================================================
// Kernel C1: per mention, top-32 of the 512 row maxima (descending, stable
// tie-break on smaller index, matching lax.top_k). One wave32 per mention.
// ============================================================
__global__ __launch_bounds__(32) void k_topk(
    const float* __restrict__ row_max, const int* __restrict__ row_arg,
    const int* __restrict__ ent_ids, int* __restrict__ top_ids,
    float* __restrict__ ent_out) {
  const int n = blockIdx.x;
  const int lane = threadIdx.x & 31;
  float v[16];
  for (int i = 0; i < 16; ++i)
    v[i] = row_max[(long)n * NROWS + lane * 16 + i];
  unsigned consumed = 0;
  for (int kk = 0; kk < KTOP; ++kk) {
    float bv = -FLT_MAX; int bi = -1;
    for (int i = 0; i < 16; ++i)
      if (!((consumed >> i) & 1u) && v[i] > bv) { bv = v[i]; bi = i; }
    int gi = (bi < 0) ? 0x7FFFFFFF : lane * 16 + bi;
    for (int off = 16; off; off >>= 1) {          // wave32 butterfly argmax
      float ov = __shfl_xor(bv, off, 32);
      int   og = __shfl_xor(gi, off, 32);
      if (ov > bv || (ov == bv && og < gi)) { bv = ov; gi = og; }
    }
    if (gi != 0x7FFFFFFF && (gi >> 4) == lane) consumed |= 1u << (gi & 15);
    if (lane == 0) {
      int loc = row_arg[(long)n * NROWS + gi];
      int id  = gi * VPR + loc;
      top_ids[n * KTOP + kk] = id;
      ent_out[n * KTOP + kk] = (float)ent_ids[id];   // < 1e6: exact in f32
    }
  }
}

// ============================================================
// Kernel C2: exact f32 rescore of the 32 selected keys, mask penalty,
// softmax, and value pooling.  One block (256 thr) per mention.
// ============================================================
__global__ __launch_bounds__(256) void k_attend(
    const float* __restrict__ qbuf, const float* __restrict__ keys,
    const float* __restrict__ vals, const int* __restrict__ mmask,
    const int* __restrict__ top_ids, float* __restrict__ attn_out,
    float* __restrict__ pooled) {
  __shared__ float qs[DKk];
  __shared__ int   ids[KTOP];
  __shared__ float sc[KTOP];
  __shared__ float wsx[KTOP];
  const int n = blockIdx.x, tid = threadIdx.x;
  if (tid < DKk)  qs[tid]  = qbuf[(long)n * DKk + tid];
  if (tid < KTOP) ids[tid] = top_ids[n * KTOP + tid];
  __syncthreads();
  {
    const int j = tid >> 3, r = tid & 7;           // 8 lanes per key
    const float* krow = keys + (long)ids[j] * DKk;
    float p = 0.f;
    for (int i = 0; i < 16; ++i) p += qs[r * 16 + i] * krow[r * 16 + i];
    p += __shfl_xor(p, 4, 8);
    p += __shfl_xor(p, 2, 8);
    p += __shfl_xor(p, 1, 8);
    if (r == 0) sc[j] = p;
  }
  __syncthreads();
  if (tid < KTOP) {
    float s = sc[tid] - (1.0f - (float)mmask[ids[tid]]) * 1e10f;
    float mx = s;
    for (int off = 16; off; off >>= 1) mx = fmaxf(mx, __shfl_xor(mx, off, 32));
    float e = __expf(s - mx);
    float sm = e;
    for (int off = 16; off; off >>= 1) sm += __shfl_xor(sm, off, 32);
    float w = e / sm;
    attn_out[n * KTOP + tid] = w;
    wsx[tid] = w;
  }
  __syncthreads();
  for (int h = tid; h < DVv; h += 256) {
    float a = 0.f;
    for (int j = 0; j < KTOP; ++j)
      a += wsx[j] * vals[(long)ids[j] * DVv + h];
    pooled[(long)n * DVv + h] = a;
  }
}

// ============================================================
// Kernel D: projected[N,768] = pooled[N,512] @ update_w + ub, * mention_mask
// Grid (N/16, 6), 256 thr: wave w owns cols [by*128 + 16w, +16).
// ============================================================
__global__ __launch_bounds__(256) void k_project(
    const float* __restrict__ pooled, const float* __restrict__ uw,
    const float* __restrict__ ub, const float* __restrict__ mmask_f,
    float* __restrict__ proj) {
  __shared__ float As[16][DVv + 4];     // 33KB, pad 4 -> conflict free
  const int tid  = threadIdx.x;
  const int lane = tid & 31, wave = tid >> 5;
  const int rbase = blockIdx.x * 16;
  const int cbase = blockIdx.y * 128 + wave * 16;
  for (int idx = tid; idx < 16 * DVv; idx += 256)
    As[idx / DVv][idx % DVv] = pooled[(long)(rbase + idx / DVv) * DVv + (idx % DVv)];
  __syncthreads();
  const int m = lane & 15, koff = (lane >> 4) * 2;
  v8f acc = {};
  for (int k = 0; k < DVv; k += 4) {
    v2f a, b;
    a.x = As[m][k + koff];
    a.y = As[m][k + koff + 1];
    b.x = uw[(long)(k + koff) * Hh + cbase + m];
    b.y = uw[(long)(k + koff + 1) * Hh + cbase + m];
    acc = wmma4(a, b, acc);
  }
  const float bias = ub[cbase + m];
  for (int v = 0; v < 8; ++v) {
    int row = v + 8 * (lane >> 4);
    proj[(long)(rbase + row) * Hh + cbase + m] =
        (acc[v] + bias) * mmask_f[rbase + row];
  }
}

// ============================================================
// Kernel E: deterministic scatter-add (position scans mentions in order,
// no float atomics) fused with LayerNorm. One block per (b,t); H = 3*256.
// Mention (b,t) keys staged once into LDS; scan is broadcast ds_loads.
// ============================================================
__global__ __launch_bounds__(256) void k_scatter_ln(
    const float* __restrict__ enc, const int* __restrict__ mbp,
    const int* __restrict__ msp, const float* __restrict__ proj,
    const float* __restrict__ gamma, const float* __restrict__ beta,
    float* __restrict__ out) {
  __shared__ int   mkey[Nn];            // 4KB: mention -> b*T + t
  __shared__ float redS[8];
  __shared__ float redQ[8];
  const int tid = threadIdx.x;
  const int pos = blockIdx.x;           // == b*Tt + t
  for (int i = tid; i < Nn; i += 256) mkey[i] = mbp[i] * Tt + msp[i];
  const long base = (long)pos * Hh;
  float x0 = enc[base + tid];
  float x1 = enc[base + tid + 256];
  float x2 = enc[base + tid + 512];
  __syncthreads();
  for (int n = 0; n < Nn; ++n) {
    if (mkey[n] == pos) {               // fixed ascending order -> deterministic
      x0 += proj[(long)n * Hh + tid];
      x1 += proj[(long)n * Hh + tid + 256];
      x2 += proj[(long)n * Hh + tid + 512];
    }
  }
  float s = x0 + x1 + x2;
  float q = x0 * x0 + x1 * x1 + x2 * x2;
  for (int off = 16; off; off >>= 1) {
    s += __shfl_xor(s, off, 32);
    q += __shfl_xor(q, off, 32);
  }
  const int wave = tid >> 5, lane = tid & 31;
  if (lane == 0) { redS[wave] = s; redQ[wave] = q; }
  __syncthreads();
  if (tid == 0) {
    float ts = 0.f, tq = 0.f;
    for (int w = 0; w < 8; ++w) { ts += redS[w]; tq += redQ[w]; }
    redS[0] = ts; redQ[0] = tq;
  }
  __syncthreads();
  const float mean = redS[0] / (float)Hh;
  const float var  = redQ[0] / (float)Hh - mean * mean;
  const float rstd = rsqrtf(var + 1e-12f);
  out[base + tid]       = (x0 - mean) * rstd * gamma[tid]       + beta[tid];
  out[base + tid + 256] = (x1 - mean) * rstd * gamma[tid + 256] + beta[tid + 256];
  out[base + tid + 512] = (x2 - mean) * rstd * gamma[tid + 512] + beta[tid + 512];
}

// ============================================================
extern "C" void kernel_launch(void* const* d_in, const int* in_sizes, int n_in,
                              void* d_out, int out_size, void* d_ws, size_t ws_size,
                              hipStream_t stream) {
  const float* enc      = (const float*)d_in[0];
  const int*   mbp      = (const int*)  d_in[1];
  const int*   msp      = (const int*)  d_in[2];
  const int*   mep      = (const int*)  d_in[3];
  const float* mmask_f  = (const float*)d_in[4];
  const float* mem_keys = (const float*)d_in[5];
  const float* mem_vals = (const float*)d_in[6];
  const int*   mem_mask = (const int*)  d_in[7];
  const int*   ent_ids  = (const int*)  d_in[8];
  const float* qw       = (const float*)d_in[9];
  const float* qb       = (const float*)d_in[10];
  const float* uw       = (const float*)d_in[11];
  const float* ub       = (const float*)d_in[12];
  const float* ln_scale = (const float*)d_in[13];
  const float* ln_bias  = (const float*)d_in[14];

  // workspace layout (floats)
  float* ws      = (float*)d_ws;
  float* q_buf   = ws;                              // N*128
  float* row_max = q_buf   + (long)Nn * DKk;        // N*512
  int*   row_arg = (int*)  (row_max + (long)Nn * NROWS);
  int*   top_ids = (int*)  ((float*)row_arg + (long)Nn * NROWS);
  float* pooled  = (float*)top_ids + (long)Nn * KTOP;
  float* proj    = pooled  + (long)Nn * DVv;

  // d_out = enc[B,T,H] ++ attn[N,K] ++ entity_ids[N,K] (as f32, exact < 2^24)
  float* out_enc  = (float*)d_out;
  float* out_attn = out_enc  + (long)Bb * Tt * Hh;
  float* out_ent  = out_attn + (long)Nn * KTOP;

  k_queries<<<dim3(Nn / 16), dim3(256), 0, stream>>>(
      enc, mbp, msp, mep, qw, qb, q_buf);
  k_scores<<<dim3(Nn / 16, NROWS), dim3(256), 0, stream>>>(
      q_buf, mem_keys, row_max, row_arg);
  k_topk<<<dim3(Nn), dim3(32), 0, stream>>>(
      row_max, row_arg, ent_ids, top_ids, out_ent);
  k_attend<<<dim3(Nn), dim3(256), 0, stream>>>(
      q_buf, mem_keys, mem_vals, mem_mask, top_ids, out_attn, pooled);
  k_project<<<dim3(Nn / 16, Hh / 128), dim3(256), 0, stream>>>(
      pooled, uw, ub, mmask_f, proj);
  k_scatter_ln<<<dim3(Bb * Tt), dim3(256), 0, stream>>>(
      enc, mbp, msp, proj, ln_scale, ln_bias, out_enc);
}